// HOA_26628797236052
// MI455X (gfx1250) — compile-verified
//
#include <hip/hip_runtime.h>
#include <hip/hip_bf16.h>

#define NN 100000
#define NE 1600000
#define D  128
#define DO3 384   // (ORDER+1)*DOUT
#define EPS 1e-9f

typedef __attribute__((ext_vector_type(2))) float v2f;
typedef __attribute__((ext_vector_type(8))) float v8f;

// -------- SpMM: y[rows[e],:] += vals[e] * x[cols[e],:]  (atomic scatter) ----
// one wave (32 lanes) per edge, 4 consecutive floats per lane
__global__ __launch_bounds__(256) void spmm_kernel(
    const int* __restrict__ rows, const int* __restrict__ cols,
    const float* __restrict__ vals, const float* __restrict__ x,
    float* __restrict__ y)
{
    const int t = blockIdx.x * 256 + threadIdx.x;
    const int e = t >> 5;
    if (e >= NE) return;
    const int c = (t & 31) * 4;

    const int   r  = rows[e];
    const int   cc = cols[e];
    const float v  = vals[e];

    const float4 xv = *(const float4*)(x + (size_t)cc * D + c);
    float* yp = y + (size_t)r * D + c;
    atomicAdd(yp + 0, v * xv.x);
    atomicAdd(yp + 1, v * xv.y);
    atomicAdd(yp + 2, v * xv.z);
    atomicAdd(yp + 3, v * xv.w);
}

// -------- fused GEMM (h @ W + b) + ReLU + per-row mean/var norm ------------
// block = 16 rows x 128 cols of output; 8 waves, each wave owns a 16-wide N strip
__global__ __launch_bounds__(256) void hop_gemm_norm(
    const float* __restrict__ h,      // [N, 128]
    const float* __restrict__ Wm,     // [128, 128] row-major (k, n)
    const float* __restrict__ bias,   // [128]
    const float* __restrict__ scale,  // [128]
    const float* __restrict__ offset, // [128]
    float* __restrict__ out,          // [N, 384]
    int colbase)                      // hop * 128
{
    __shared__ float Ash[16][132];    // padded: stride 132 -> conflict-free A reads
    __shared__ float fsh[16][128];
    __shared__ float psum[16][16];
    __shared__ float psq [16][16];
    __shared__ float rmean[16];
    __shared__ float rinv [16];

    const int tid  = threadIdx.x;
    const int row0 = blockIdx.x * 16;     // N = 100000 = 6250 * 16, exact

    // stage A tile [16 x 128] into LDS (coalesced)
    for (int idx = tid; idx < 16 * D; idx += 256) {
        const int r = idx >> 7, c = idx & 127;
        Ash[r][c] = h[(size_t)(row0 + r) * D + c];
    }
    __syncthreads();

    const int wave = tid >> 5;
    const int lane = tid & 31;
    const int half = lane >> 4;           // 0: lanes 0-15, 1: lanes 16-31
    const int lm   = lane & 15;
    const int n0   = wave * 16;
    const int col  = n0 + lm;             // this lane's output column

    // preload B fragments for the whole K=128 strip (stays in VGPRs)
    // B 4x16 tile: VGPR0 = (K = k0+2*half, N = lm), VGPR1 = (K+1, N = lm)
    v2f bfrag[32];
#pragma unroll
    for (int kk = 0; kk < 32; ++kk) {
        const int k = kk * 4 + half * 2;
        bfrag[kk].x = Wm[(size_t)k * D + col];
        bfrag[kk].y = Wm[(size_t)(k + 1) * D + col];
    }

    // C initialized with bias (bias is per-column, identical for all 16 rows)
    const float bv = bias[col];
    v8f acc = { bv, bv, bv, bv, bv, bv, bv, bv };

    // 32 x v_wmma_f32_16x16x4_f32 : full-precision fp32 matrix pipe
#pragma unroll
    for (int kk = 0; kk < 32; ++kk) {
        const int k = kk * 4 + half * 2;
        v2f a;
        a.x = Ash[lm][k];                 // A 16x4: lane lm = row M, K = k .. k+1
        a.y = Ash[lm][k + 1];
        acc = __builtin_amdgcn_wmma_f32_16x16x4_f32(
            false, a, false, bfrag[kk], (short)0, acc, false, false);
    }

    // ReLU + stage f tile into LDS (C/D layout: VGPR r -> M = r + 8*half, N = lm)
#pragma unroll
    for (int r = 0; r < 8; ++r) {
        float f = acc[r];
        f = f > 0.0f ? f : 0.0f;
        fsh[r + 8 * half][col] = f;
    }
    __syncthreads();

    // per-row mean / var: stage 1 — 16 partials of 8 columns each
    {
        const int r = tid >> 4, seg = tid & 15;
        float s = 0.0f, sq = 0.0f;
#pragma unroll
        for (int j = 0; j < 8; ++j) {
            const float f = fsh[r][seg * 8 + j];
            s += f; sq += f * f;
        }
        psum[r][seg] = s; psq[r][seg] = sq;
    }
    __syncthreads();

    // stage 2 — one thread per row
    if (tid < 16) {
        float s = 0.0f, sq = 0.0f;
#pragma unroll
        for (int j = 0; j < 16; ++j) { s += psum[tid][j]; sq += psq[tid][j]; }
        const float mean = s * (1.0f / 128.0f);
        const float var  = sq * (1.0f / 128.0f) - mean * mean + EPS;
        rmean[tid] = mean;
        rinv[tid]  = rsqrtf(var);
    }
    __syncthreads();

    // normalize + write into strided [N, 384] output
    for (int idx = tid; idx < 16 * D; idx += 256) {
        const int r = idx >> 7, c = idx & 127;
        const float f = fsh[r][c];
        out[(size_t)(row0 + r) * DO3 + colbase + c] =
            (f - rmean[r]) * scale[c] * rinv[r] + offset[c];
    }
}

extern "C" void kernel_launch(void* const* d_in, const int* in_sizes, int n_in,
                              void* d_out, int out_size, void* d_ws, size_t ws_size,
                              hipStream_t stream) {
    const int*   rows   = (const int*)  d_in[0];
    const int*   cols   = (const int*)  d_in[1];
    const float* vals   = (const float*)d_in[2];
    const float* feat   = (const float*)d_in[3];
    const float* W      = (const float*)d_in[4];   // [3,128,128]
    const float* b      = (const float*)d_in[5];   // [3,128]
    const float* scale  = (const float*)d_in[6];   // [3,128]
    const float* offset = (const float*)d_in[7];   // [3,128]
    float*       out    = (float*)d_out;           // [N,384]

    float* h1 = (float*)d_ws;
    float* h2 = h1 + (size_t)NN * D;

    // zero the SpMM accumulators (capturable async memset)
    hipMemsetAsync(d_ws, 0, 2 * (size_t)NN * D * sizeof(float), stream);

    const int spmmBlocks = NE / 8;   // 8 edges per 256-thread block
    spmm_kernel<<<spmmBlocks, 256, 0, stream>>>(rows, cols, vals, feat, h1);
    spmm_kernel<<<spmmBlocks, 256, 0, stream>>>(rows, cols, vals, h1,   h2);

    const int gemmBlocks = NN / 16;  // 6250
    hop_gemm_norm<<<gemmBlocks, 256, 0, stream>>>(feat, W,           b,         scale,         offset,         out, 0);
    hop_gemm_norm<<<gemmBlocks, 256, 0, stream>>>(h1,   W + D * D,   b + D,     scale + D,     offset + D,     out, D);
    hop_gemm_norm<<<gemmBlocks, 256, 0, stream>>>(h2,   W + 2 * D * D, b + 2 * D, scale + 2 * D, offset + 2 * D, out, 2 * D);
}